// BinTreeNetwork_35450660061315
// MI455X (gfx1250) — compile-verified
//
#include <hip/hip_runtime.h>

// ---------------------------------------------------------------------------
// BinTreeNetwork on gfx1250 (MI455X).
// out[a, j20..j1, o] = c[o] + Wo0·u(j1) + sum_{i=1..19} Wo_i·s_i^{(j_{i+1})}
//                    + Wo20·s_20^{(a)}
// HBM-store-bound (256 MB out); tree expanded hierarchically; the 32-wide
// output accumulation runs on V_WMMA_F32_16X16X4_F32.
// This revision: (a) hoists the Wo B-operands out of the per-group loop,
// (b) rewrites stores as saddr + 32-bit voffset + imm24 so the store clause
// needs no per-store 64-bit address math.
// ---------------------------------------------------------------------------

typedef float v2f __attribute__((ext_vector_type(2)));
typedef float v8f __attribute__((ext_vector_type(8)));

#define IN_DIM 4096
#define OUTD 32
#define DSPLIT 10                 // prefix depth handled by phase B
#define NPREFIX (1 << DSPLIT)     // 1024 blocks in phase C
#define RLEV 10                   // remaining levels (10..19), +final level 20
#define NLEAF (1 << RLEV)         // 1024 leaves per phase-C block

// workspace float offsets
#define WS_L0 0                   // left0[2]
#define WS_R0 2                   // right0[2]
#define WS_C 4                    // c[32] = x@out_layer0^T + out_bias
#define WS_PAIR0 36
#define WS_PAIR1 (36 + 4 * NPREFIX)
#define WS_ACC0 (36 + 8 * NPREFIX)
#define WS_ACC1 (36 + 8 * NPREFIX + 32 * NPREFIX)

// ------------------------- Phase A: input projections -----------------------
__global__ __launch_bounds__(256) void bintree_phaseA(
    const float* __restrict__ x,
    const float* __restrict__ inL, const float* __restrict__ inR,
    const float* __restrict__ bL, const float* __restrict__ bR,
    const float* __restrict__ outW0, const float* __restrict__ outBias,
    float* __restrict__ ws) {
  __shared__ float red[256];
  const int r = blockIdx.x;  // 0..35
  const float* wrow;
  if (r < 2)
    wrow = inL + r * IN_DIM;
  else if (r < 4)
    wrow = inR + (r - 2) * IN_DIM;
  else
    wrow = outW0 + (r - 4) * IN_DIM;
  float s = 0.f;
  for (int j = threadIdx.x; j < IN_DIM; j += 256) s += x[j] * wrow[j];
  red[threadIdx.x] = s;
  __syncthreads();
  for (int off = 128; off > 0; off >>= 1) {
    if (threadIdx.x < off) red[threadIdx.x] += red[threadIdx.x + off];
    __syncthreads();
  }
  if (threadIdx.x == 0) {
    float v = red[0];
    if (r < 2)
      ws[WS_L0 + r] = v + bL[r];
    else if (r < 4)
      ws[WS_R0 + (r - 2)] = v + bR[r - 2];
    else
      ws[WS_C + (r - 4)] = v + outBias[r - 4];
  }
}

// --------------- Phase B: prefix tree to depth DSPLIT (1 block) -------------
// per prefix p (bits j1..j10, j1 = bit9, j10 = bit0):
//   pairs[p] = {s^{(0)}.x, s^{(0)}.y, s^{(1)}.x, s^{(1)}.y}   (level-10 states)
//   acc[p][o] = c[o] + Wo0·u(j1) + sum_{k=1..9} Wo_k·s_k^{(j_{k+1})}
__global__ __launch_bounds__(256) void bintree_phaseB(
    const float* __restrict__ tL, const float* __restrict__ tR,
    const float* __restrict__ tBL, const float* __restrict__ tBR,
    const float* __restrict__ outW,  // (21,32,2)
    float* __restrict__ ws) {
  const float* left0 = ws + WS_L0;
  const float* right0 = ws + WS_R0;
  const float* cvec = ws + WS_C;
  float* pairs[2] = {ws + WS_PAIR0, ws + WS_PAIR1};
  float* acc[2] = {ws + WS_ACC0, ws + WS_ACC1};
  const int tid = threadIdx.x;

  // level 1 (2 prefixes)
  if (tid < 2) {
    const int j1 = tid;
    const float* M = (j1 ? tR : tL);   // layer 0
    const float* B = (j1 ? tBR : tBL);
    for (int a0 = 0; a0 < 2; ++a0) {
      const float ux = left0[a0], uy = right0[a0];
      pairs[0][j1 * 4 + a0 * 2 + 0] = M[0] * ux + M[1] * uy + B[0];
      pairs[0][j1 * 4 + a0 * 2 + 1] = M[2] * ux + M[3] * uy + B[1];
    }
  }
  if (tid < 64) {
    const int j1 = tid >> 5, o = tid & 31;
    acc[0][j1 * 32 + o] =
        cvec[o] + outW[o * 2 + 0] * left0[j1] + outW[o * 2 + 1] * right0[j1];
  }
  __syncthreads();

  int cur = 0;
  for (int i = 1; i < DSPLIT; ++i) {  // produce level i+1 using layer i
    const int nch = 1 << (i + 1);
    const float* pin = pairs[cur];
    float* pout = pairs[cur ^ 1];
    const float* ain = acc[cur];
    float* aout = acc[cur ^ 1];
    for (int c = tid; c < nch; c += 256) {
      const int par = c >> 1, b = c & 1;
      const float* M = (b ? tR : tL) + i * 4;
      const float* B = (b ? tBR : tBL) + i * 2;
      const float p0 = pin[par * 4 + 0], p1 = pin[par * 4 + 1];
      const float p2 = pin[par * 4 + 2], p3 = pin[par * 4 + 3];
      pout[c * 4 + 0] = M[0] * p0 + M[1] * p1 + B[0];
      pout[c * 4 + 1] = M[2] * p0 + M[3] * p1 + B[1];
      pout[c * 4 + 2] = M[0] * p2 + M[1] * p3 + B[0];
      pout[c * 4 + 3] = M[2] * p2 + M[3] * p3 + B[1];
    }
    for (int t = tid; t < nch * 32; t += 256) {
      const int c = t >> 5, o = t & 31;
      const int par = c >> 1, b = c & 1;
      const float sx = pin[par * 4 + b * 2 + 0];
      const float sy = pin[par * 4 + b * 2 + 1];
      aout[c * 32 + o] =
          ain[par * 32 + o] + outW[i * 64 + o * 2] * sx + outW[i * 64 + o * 2 + 1] * sy;
    }
    __syncthreads();
    cur ^= 1;
  }
}

// -------- Phase C: per-prefix subtree + WMMA output (1024 blocks) -----------
__global__ __launch_bounds__(256) void bintree_phaseC(
    const float* __restrict__ tL, const float* __restrict__ tR,
    const float* __restrict__ tBL, const float* __restrict__ tBR,
    const float* __restrict__ outW,      // (21,32,2)
    const float* __restrict__ pairRoot,  // NPREFIX*4
    const float* __restrict__ accRoot,   // NPREFIX*32
    float* __restrict__ out) {
  __shared__ float4 states[2 * NLEAF - 1];  // all levels d=0..10, 32 KB
  const int tid = threadIdx.x;
  const unsigned prefix = blockIdx.x;

  __builtin_prefetch(accRoot + prefix * 32, 0, 3);

  if (tid == 0) {
    const float* p = pairRoot + prefix * 4;
    states[0] = make_float4(p[0], p[1], p[2], p[3]);
  }
  __syncthreads();

  // BFS states for abs levels 11..20 using tree layers 10..19
  for (int d = 0; d < RLEV; ++d) {
    const int i = DSPLIT + d;
    const int nch = 1 << (d + 1);
    const int baseP = (1 << d) - 1;
    const int baseC = (1 << (d + 1)) - 1;
    for (int c = tid; c < nch; c += 256) {
      const int par = c >> 1, b = c & 1;
      const float* M = (b ? tR : tL) + i * 4;
      const float* B = (b ? tBR : tBL) + i * 2;
      const float4 p = states[baseP + par];
      float4 q;
      q.x = M[0] * p.x + M[1] * p.y + B[0];
      q.y = M[2] * p.x + M[3] * p.y + B[1];
      q.z = M[0] * p.z + M[1] * p.w + B[0];
      q.w = M[2] * p.z + M[3] * p.w + B[1];
      states[baseC + c] = q;
    }
    __syncthreads();
  }

  // Output: 16 leaves per WMMA group; A 16x4 f32 (rows=leaf, K packs two
  // levels), B 4x16 f32 (Wo columns), C 16x16 f32 (rows=leaf, cols=o-half).
  const int lane = tid & 31;
  const int wave = tid >> 5;
  const int m = lane & 15;       // A-row / B-col / C-col within tile
  const int hiHalf = lane >> 4;  // 0: K=0..1 (level d), 1: K=2..3 (level d+1)
  const float* accP = accRoot + prefix * 32;
  const float accLo = accP[m];
  const float accHi = accP[16 + m];

  // Hoist Wo operands: per dpair dp, this lane supplies level DSPLIT+2*dp+hiHalf.
  const float2* outW2 = (const float2*)outW;  // (21,32) float2
  v2f Bh0[RLEV / 2], Bh1[RLEV / 2];
#pragma unroll
  for (int dp = 0; dp < RLEV / 2; ++dp) {
    const int lvl = DSPLIT + 2 * dp + hiHalf;  // 10..19
    const float2 w0 = outW2[lvl * 32 + m];
    const float2 w1 = outW2[lvl * 32 + 16 + m];
    Bh0[dp].x = w0.x; Bh0[dp].y = w0.y;
    Bh1[dp].x = w1.x; Bh1[dp].y = w1.y;
  }
  v2f B20a = {0.f, 0.f}, B20b = {0.f, 0.f};
  if (hiHalf == 0) {
    const float2 wa = outW2[20 * 32 + m];
    const float2 wb = outW2[20 * 32 + 16 + m];
    B20a.x = wa.x; B20a.y = wa.y;
    B20b.x = wb.x; B20b.y = wb.y;
  }

  for (int g = wave; g < NLEAF / 16; g += 8) {
    const int leafBase = g << 4;
    const int leaf = leafBase + m;
    v8f c0, c1;
#pragma unroll
    for (int r = 0; r < 8; ++r) { c0[r] = accLo; c1[r] = accHi; }

#pragma unroll
    for (int dp = 0; dp < RLEV / 2; ++dp) {
      const int dd = 2 * dp + hiHalf;  // this lane supplies level DSPLIT+dd
      const int idx = leaf & ((1 << dd) - 1);
      const int var = (leaf >> dd) & 1;
      const float4 nd = states[((1 << dd) - 1) + idx];
      v2f A;
      A.x = var ? nd.z : nd.x;
      A.y = var ? nd.w : nd.y;
      c0 = __builtin_amdgcn_wmma_f32_16x16x4_f32(false, A, false, Bh0[dp],
                                                 (short)0, c0, false, false);
      c1 = __builtin_amdgcn_wmma_f32_16x16x4_f32(false, A, false, Bh1[dp],
                                                 (short)0, c1, false, false);
    }

    // final level-20 term, per a-variant
    const float4 leafn = states[(NLEAF - 1) + leaf];
    v2f A0 = {0.f, 0.f}, A1 = {0.f, 0.f};
    if (hiHalf == 0) {
      A0.x = leafn.x; A0.y = leafn.y;
      A1.x = leafn.z; A1.y = leafn.w;
    }
    const v8f d00 = __builtin_amdgcn_wmma_f32_16x16x4_f32(false, A0, false, B20a,
                                                          (short)0, c0, false, false);
    const v8f d01 = __builtin_amdgcn_wmma_f32_16x16x4_f32(false, A0, false, B20b,
                                                          (short)0, c1, false, false);
    const v8f d10 = __builtin_amdgcn_wmma_f32_16x16x4_f32(false, A1, false, B20a,
                                                          (short)0, c0, false, false);
    const v8f d11 = __builtin_amdgcn_wmma_f32_16x16x4_f32(false, A1, false, B20b,
                                                          (short)0, c1, false, false);

    // Stores: saddr(out) + 32-bit voffset + imm24.
    // float index = prefix*32 + leaf*32768 (+ r*32768) + half*16 + m (+ a*(32<<20))
    const unsigned base0 =
        (prefix << 5) + ((unsigned)(leafBase + 8 * hiHalf) << 15) + (unsigned)m;
    const unsigned base1 = base0 + (32u << 20);  // a = 1 half
    const size_t b0 = (size_t)base0;
    const size_t b1 = (size_t)base1;
#pragma unroll
    for (int r = 0; r < 8; ++r) {
      const size_t roff = (size_t)r << 15;  // folds into imm24 (r*128KiB bytes)
      out[b0 + roff] = d00[r];
      out[b0 + roff + 16] = d01[r];
      out[b1 + roff] = d10[r];
      out[b1 + roff + 16] = d11[r];
    }
  }
}

// ------------------------------- launcher -----------------------------------
extern "C" void kernel_launch(void* const* d_in, const int* in_sizes, int n_in,
                              void* d_out, int out_size, void* d_ws, size_t ws_size,
                              hipStream_t stream) {
  const float* x   = (const float*)d_in[0];
  const float* inL = (const float*)d_in[1];
  const float* inR = (const float*)d_in[2];
  const float* bL  = (const float*)d_in[3];
  const float* bR  = (const float*)d_in[4];
  const float* tL  = (const float*)d_in[5];
  const float* tR  = (const float*)d_in[6];
  const float* tBL = (const float*)d_in[7];
  const float* tBR = (const float*)d_in[8];
  const float* oW0 = (const float*)d_in[9];
  const float* oW  = (const float*)d_in[10];
  const float* oB  = (const float*)d_in[11];
  float* ws = (float*)d_ws;
  float* out = (float*)d_out;

  bintree_phaseA<<<36, 256, 0, stream>>>(x, inL, inR, bL, bR, oW0, oB, ws);
  bintree_phaseB<<<1, 256, 0, stream>>>(tL, tR, tBL, tBR, oW, ws);

  // DSPLIT-1 = 9 ping-pong transitions -> final buffers are index 1
  const int fin = (DSPLIT - 1) & 1;
  const float* pairFinal = ws + (fin ? WS_PAIR1 : WS_PAIR0);
  const float* accFinal = ws + (fin ? WS_ACC1 : WS_ACC0);
  bintree_phaseC<<<NPREFIX, 256, 0, stream>>>(tL, tR, tBL, tBR, oW, pairFinal,
                                              accFinal, out);
}